// ConstrativeLoss_71124658422192
// MI455X (gfx1250) — compile-verified
//
#include <hip/hip_runtime.h>
#include <hip/hip_bf16.h>
#include <math.h>

// Problem constants (from reference): x is [16384, 512] f32, T = 0.5, EPS = 1e-8.
#define N_ROWS 16384
#define N_COLS 512
#define EPSV   1e-8f

typedef __attribute__((ext_vector_type(2))) float v2f;
typedef __attribute__((ext_vector_type(8))) float v8f;

__device__ __forceinline__ float wave_reduce_sum(float v) {
#pragma unroll
    for (int m = 16; m >= 1; m >>= 1) v += __shfl_xor(v, m, 32);
    return v;
}

// ---------------------------------------------------------------------------
// Pass 1: stream x once (32 MB). Each wave handles whole rows:
//   lane l owns columns [16l, 16l+16) loaded as 4x float4 (global_load_b128).
// Per row: sumsq -> wave butterfly -> norm clamp -> scale-accumulate into
// per-lane register column accumulators + diag partial.
// Deterministic LDS combine (no float atomics), then write block partials.
// ---------------------------------------------------------------------------
__global__ __launch_bounds__(256) void contrastive_pass1(
    const float* __restrict__ x,
    float* __restrict__ partial,   // [nBlocks][512] column partial sums
    float* __restrict__ diagp,     // [nBlocks]
    int rowsPerBlock)              // must be multiple of 8 (waves per block)
{
    const int lane = threadIdx.x & 31;
    const int wave = threadIdx.x >> 5;      // 0..7
    const int rowsPerWave = rowsPerBlock >> 3;

    float acc[16];
#pragma unroll
    for (int k = 0; k < 16; ++k) acc[k] = 0.0f;
    float dacc = 0.0f;

    const int row0 = blockIdx.x * rowsPerBlock + wave * rowsPerWave;

    for (int r = 0; r < rowsPerWave; ++r) {
        const float* xr = x + (size_t)(row0 + r) * N_COLS + lane * 16;
        float4 v0 = *(const float4*)(xr + 0);
        float4 v1 = *(const float4*)(xr + 4);
        float4 v2 = *(const float4*)(xr + 8);
        float4 v3 = *(const float4*)(xr + 12);

        float ss = v0.x * v0.x + v0.y * v0.y + v0.z * v0.z + v0.w * v0.w
                 + v1.x * v1.x + v1.y * v1.y + v1.z * v1.z + v1.w * v1.w
                 + v2.x * v2.x + v2.y * v2.y + v2.z * v2.z + v2.w * v2.w
                 + v3.x * v3.x + v3.y * v3.y + v3.z * v3.z + v3.w * v3.w;
        ss = wave_reduce_sum(ss);                      // identical on all lanes

        float nrm = fmaxf(sqrtf(ss), EPSV);           // torchmetrics eps clamp
        float inv = 1.0f / nrm;
        dacc += ss * inv * inv;                        // n_i . n_i (diag term)

        acc[0]  += v0.x * inv;  acc[1]  += v0.y * inv;
        acc[2]  += v0.z * inv;  acc[3]  += v0.w * inv;
        acc[4]  += v1.x * inv;  acc[5]  += v1.y * inv;
        acc[6]  += v1.z * inv;  acc[7]  += v1.w * inv;
        acc[8]  += v2.x * inv;  acc[9]  += v2.y * inv;
        acc[10] += v2.z * inv;  acc[11] += v2.w * inv;
        acc[12] += v3.x * inv;  acc[13] += v3.y * inv;
        acc[14] += v3.z * inv;  acc[15] += v3.w * inv;
    }

    // Deterministic block combine: per-wave slots in LDS, fixed-order sum.
    __shared__ float sblk[8][N_COLS];    // 16 KB (of 320 KB LDS/WGP)
    __shared__ float dblk[8];
#pragma unroll
    for (int k = 0; k < 16; ++k) sblk[wave][lane * 16 + k] = acc[k];
    if (lane == 0) dblk[wave] = dacc;
    __syncthreads();

    for (int c = threadIdx.x; c < N_COLS; c += 256) {
        float s = 0.0f;
#pragma unroll
        for (int w = 0; w < 8; ++w) s += sblk[w][c];
        partial[(size_t)blockIdx.x * N_COLS + c] = s;
    }
    if (threadIdx.x == 0) {
        float ds = 0.0f;
#pragma unroll
        for (int w = 0; w < 8; ++w) ds += dblk[w];
        diagp[blockIdx.x] = ds;
    }
}

// ---------------------------------------------------------------------------
// Pass 2: s[c] = sum over blocks of partial[b][c]  == ones^T @ P, computed with
// v_wmma_f32_16x16x4_f32 (A = ones 16x4). With A all-ones the K-slot layout of
// B is irrelevant (summed over), we only rely on N = lane&15 in B and the
// documented D layout (VGPR0, lanes 0-15 => row M=0, N=lane). f32 end-to-end.
// Then result = (||s||^2 - sum(diag)) / (T*N).
// ---------------------------------------------------------------------------
__global__ __launch_bounds__(512) void contrastive_pass2(
    const float* __restrict__ partial,
    const float* __restrict__ diagp,
    float* __restrict__ out,
    int nBlocks)
{
    const int lane = threadIdx.x & 31;
    const int wave = threadIdx.x >> 5;       // 0..15
    __shared__ float s_sh[N_COLS];

    v2f ones;
    ones.x = 1.0f; ones.y = 1.0f;
    const int half = lane >> 4;              // 0 or 1
    const int n    = lane & 15;              // output column within tile

    // 32 tiles of 16 columns over 16 waves -> 2 tiles per wave, no divergence
    // (EXEC all-ones around every WMMA, as the ISA requires).
    for (int t = wave; t < (N_COLS / 16); t += 16) {
        const int c0 = t * 16;
        v8f cacc = {};
        for (int k = 0; k < nBlocks; k += 4) {
            const float* p = partial + (size_t)(k + half * 2) * N_COLS + c0 + n;
            v2f b;
            b.x = p[0];
            b.y = p[N_COLS];
#if defined(__AMDGCN__)
            cacc = __builtin_amdgcn_wmma_f32_16x16x4_f32(
                false, ones, false, b, (short)0, cacc, false, false);
#else
            cacc[0] += b.x + b.y;            // host-pass placeholder only
#endif
        }
        if (lane < 16) s_sh[c0 + n] = cacc[0];   // D[0][n] = column sum
    }
    __syncthreads();

    // ||s||^2 minus diag partials, 512-value deterministic block reduction.
    const int t0 = threadIdx.x;
    float v = s_sh[t0] * s_sh[t0];
    if (t0 < nBlocks) v -= diagp[t0];
    v = wave_reduce_sum(v);

    __shared__ float wsum[16];
    if (lane == 0) wsum[wave] = v;
    __syncthreads();
    if (threadIdx.x == 0) {
        float tot = 0.0f;
#pragma unroll
        for (int w = 0; w < 16; ++w) tot += wsum[w];
        out[0] = tot * (1.0f / (0.5f * (float)N_ROWS));   // /(T*N)
    }
}

extern "C" void kernel_launch(void* const* d_in, const int* in_sizes, int n_in,
                              void* d_out, int out_size, void* d_ws, size_t ws_size,
                              hipStream_t stream)
{
    const float* x = (const float*)d_in[0];
    float* out = (float*)d_out;

    // Workspace: nBlocks*512 partial floats + nBlocks diag floats.
    int nBlocks = 256;                                    // ~525 KB workspace
    if (ws_size < (size_t)(nBlocks * N_COLS + nBlocks) * sizeof(float))
        nBlocks = 64;                                     // ~132 KB fallback
    const int rowsPerBlock = N_ROWS / nBlocks;            // multiple of 8

    float* partial = (float*)d_ws;
    float* diagp   = partial + (size_t)nBlocks * N_COLS;

    contrastive_pass1<<<nBlocks, 256, 0, stream>>>(x, partial, diagp, rowsPerBlock);
    contrastive_pass2<<<1, 512, 0, stream>>>(partial, diagp, out, nBlocks);
}